// GroupFPS_73512660238513
// MI455X (gfx1250) — compile-verified
//
#include <hip/hip_runtime.h>

typedef float v2f __attribute__((ext_vector_type(2)));
typedef float v8f __attribute__((ext_vector_type(8)));
typedef int   vsi2 __attribute__((vector_size(8)));   // matches async b64 builtin

#define BATCH  32
#define NPTS   16384
#define KC     256     // centers per batch = N / 64
#define PS     64      // neighbors per center
#define CT     16      // centers per WMMA tile (M)
#define CHUNK  512     // points per LDS chunk
#define NCHUNK (NPTS / CHUNK)

#if __has_builtin(__builtin_amdgcn_global_load_async_to_lds_b64) && \
    __has_builtin(__builtin_amdgcn_s_wait_asynccnt)
#define USE_ASYNC 1
#else
#define USE_ASYNC 0
#endif

// Monotone float -> uint mapping (preserves < ordering, incl. negatives).
__device__ __forceinline__ unsigned ordkey(float f) {
  unsigned u = __float_as_uint(f);
  return u ^ ((unsigned)((int)u >> 31) | 0x80000000u);
}

// Stage one 512-point chunk (6144 B) of x into an LDS buffer.
// 768 x 8-byte units, 3 per thread (uniform -> clean per-wave ASYNCcnt).
__device__ __forceinline__ void stage_chunk(const float* __restrict__ xb,
                                            float* dstbuf, int pbase, int tid) {
#if USE_ASYNC
  const char* g = (const char*)(xb + (size_t)pbase * 3);
  char* l = (char*)dstbuf;
#pragma unroll
  for (int k = 0; k < 3; ++k) {
    const int off = (tid + k * 256) * 8;
    __builtin_amdgcn_global_load_async_to_lds_b64(
        (__attribute__((address_space(1))) vsi2*)(g + off),
        (__attribute__((address_space(3))) vsi2*)(l + off),
        /*offset=*/0, /*cpol=*/0);
  }
#else
  const unsigned long long* g =
      (const unsigned long long*)(xb + (size_t)pbase * 3);
  unsigned long long* l = (unsigned long long*)dstbuf;
#pragma unroll
  for (int k = 0; k < 3; ++k) l[tid + k * 256] = g[tid + k * 256];
#endif
}

// ---------------------------------------------------------------------------
// Kernel 1: farthest point sampling. One workgroup per batch, 1024 threads.
// min_d lives in registers (16 pts/thread); 255 dependent argmax reductions.
// Writes centers (gathered points) directly into the c region of d_out.
// ---------------------------------------------------------------------------
__global__ __launch_bounds__(1024) void fps_kernel(const float* __restrict__ x,
                                                   float* __restrict__ c_out) {
  const int b    = blockIdx.x;
  const int tid  = threadIdx.x;
  const int lane = tid & 31;
  const int wave = tid >> 5;
  const float* xb = x + (size_t)b * NPTS * 3;

  float mind[16];
#pragma unroll
  for (int j = 0; j < 16; ++j) mind[j] = 3.4e38f;

  __shared__ float s_last[3];
  __shared__ float rv[32];
  __shared__ int   ri[32];
  __shared__ int   s_next;

  int last = 0;
  for (int k = 0; k < KC; ++k) {
    if (tid == 0) {
      float lx = xb[last * 3 + 0], ly = xb[last * 3 + 1], lz = xb[last * 3 + 2];
      float* co = c_out + ((size_t)b * KC + k) * 3;
      co[0] = lx; co[1] = ly; co[2] = lz;
      s_last[0] = lx; s_last[1] = ly; s_last[2] = lz;
    }
    __syncthreads();
    if (k == KC - 1) break;

    const float lx = s_last[0], ly = s_last[1], lz = s_last[2];
    float best = -1.0f;
    int   bi   = 0;
#pragma unroll
    for (int j = 0; j < 16; ++j) {
      int i = tid + j * 1024;                       // ascending i -> stable ties
      float dx = xb[i * 3 + 0] - lx;
      float dy = xb[i * 3 + 1] - ly;
      float dz = xb[i * 3 + 2] - lz;
      float d  = dx * dx + dy * dy + dz * dz;
      float m  = mind[j] < d ? mind[j] : d;
      mind[j]  = m;
      if (m > best) { best = m; bi = i; }
    }
    // wave32 argmax reduce, ties -> lower index (matches jnp.argmax)
#pragma unroll
    for (int s = 16; s > 0; s >>= 1) {
      float ov = __shfl_down(best, s, 32);
      int   oi = __shfl_down(bi, s, 32);
      if (ov > best || (ov == best && oi < bi)) { best = ov; bi = oi; }
    }
    if (lane == 0) { rv[wave] = best; ri[wave] = bi; }
    __syncthreads();
    if (wave == 0) {
      float v  = rv[lane];
      int   i2 = ri[lane];
#pragma unroll
      for (int s = 16; s > 0; s >>= 1) {
        float ov = __shfl_down(v, s, 32);
        int   oi = __shfl_down(i2, s, 32);
        if (ov > v || (ov == v && oi < i2)) { v = ov; i2 = oi; }
      }
      if (lane == 0) s_next = i2;
    }
    __syncthreads();
    last = s_next;
  }
}

// ---------------------------------------------------------------------------
// Kernel 2: exact KNN top-64 per center via WMMA distance keys.
// One workgroup per (batch, 16-center tile): 32*16 = 512 WGs, 256 threads.
// key[m][n] = |x_n|^2 - 2 c_m . x_n  (same ranking as full d2), computed as a
// 16x16x4 f32 WMMA: A row = [-2cx,-2cy,-2cz,1], B col = [x,y,z,|x|^2].
// Point chunks are double-buffered in LDS via async global->LDS copies
// (ASYNCcnt), overlapping the next chunk's fetch with this chunk's selection.
// Keys live in LDS as ordered uints; each wave streams a sorted top-64 list
// of packed (key,idx) for 2 centers (wave-synchronous, no block barriers).
// ---------------------------------------------------------------------------
__global__ __launch_bounds__(256) void knn_kernel(const float* __restrict__ x,
                                                  const float* __restrict__ c_in,
                                                  float* __restrict__ p_out) {
  const int b     = blockIdx.x >> 4;
  const int ct    = blockIdx.x & 15;
  const int cbase = ct * CT;
  const int tid   = threadIdx.x;
  const int lane  = tid & 31;
  const int wave  = tid >> 5;
  const int hi    = lane >> 4;   // 0: lanes 0-15, 1: lanes 16-31
  const int lm16  = lane & 15;

  const float* xb = x + (size_t)b * NPTS * 3;

  __shared__ unsigned           skey[CT * CHUNK];   // 32 KB ordered keys
  __shared__ unsigned long long list[CT * PS];      //  8 KB packed (key<<32)|idx
  __shared__ float              spts[2][CHUNK * 3]; // 12 KB staged points

  for (int t = tid; t < CT * PS; t += 256) list[t] = ~0ull;

  // A operand (constant over chunks): row m = lane&15 -> [-2cx,-2cy,-2cz,1]
  // ISA A 16x4 layout: VGPR0 = K{0|2}, VGPR1 = K{1|3} split by lane half.
  const float* cp = c_in + ((size_t)b * KC + cbase + lm16) * 3;
  const float cx = cp[0], cy = cp[1], cz = cp[2];
  v2f a;
  a.x = hi ? (-2.0f * cz) : (-2.0f * cx);
  a.y = hi ? 1.0f         : (-2.0f * cy);

  stage_chunk(xb, spts[0], 0, tid);                // prologue: chunk 0

  for (int chunk = 0; chunk < NCHUNK; ++chunk) {
    const int pbase = chunk * CHUNK;
    if (chunk + 1 < NCHUNK) {
      stage_chunk(xb, spts[(chunk + 1) & 1], pbase + CHUNK, tid);
#if USE_ASYNC
      __builtin_amdgcn_s_wait_asynccnt(3);         // chunk's 3 loads landed,
#endif                                             // next chunk's 3 in flight
    } else {
#if USE_ASYNC
      __builtin_amdgcn_s_wait_asynccnt(0);
#endif
    }
    __syncthreads();                               // staged data visible to WG

    // ---- WMMA: 16 centers x 512 points of selection keys ----
    const float* sp = spts[chunk & 1];
    for (int t = wave; t < CHUNK / 16; t += 8) {
      const int nloc = t * 16 + lm16;
      const float px = sp[nloc * 3 + 0];
      const float py = sp[nloc * 3 + 1];
      const float pz = sp[nloc * 3 + 2];
      const float q  = px * px + py * py + pz * pz;
      v2f bb;                                   // B col n = [x, y, z, |x|^2]
      bb.x = hi ? pz : px;                      // rows K{0|2}
      bb.y = hi ? q  : py;                      // rows K{1|3}
      v8f acc = {};
      acc = __builtin_amdgcn_wmma_f32_16x16x4_f32(
          /*neg_a=*/false, a, /*neg_b=*/false, bb,
          /*c_mod=*/(short)0, acc, /*reuse_a=*/false, /*reuse_b=*/false);
#pragma unroll
      for (int v = 0; v < 8; ++v) {
        const int m = v + (hi << 3);            // D: M = vgpr + 8*(lane>=16)
        skey[m * CHUNK + nloc] = ordkey(acc[v]);
      }
    }
    __syncthreads();

    // ---- per-wave streaming exact top-64 merge (centers wave, wave+8) ----
#pragma unroll
    for (int gi = 0; gi < 2; ++gi) {
      const int g    = wave + gi * 8;
      const int goff = g * PS;
      while (true) {
        const unsigned long long lmax = list[goff + PS - 1];  // LDS broadcast
        unsigned long long bestp = ~0ull;
#pragma unroll
        for (int j = 0; j < CHUNK / 32; ++j) {
          const int nloc = lane + j * 32;       // 32 distinct LDS banks
          const unsigned key = skey[g * CHUNK + nloc];
          const unsigned long long pk =
              ((unsigned long long)key << 32) | (unsigned)(pbase + nloc);
          if (pk < bestp) bestp = pk;
        }
#pragma unroll
        for (int s = 16; s > 0; s >>= 1) {
          const unsigned long long o = __shfl_down(bestp, s, 32);
          if (o < bestp) bestp = o;
        }
        bestp = __shfl(bestp, 0, 32);           // uniform across wave
        if (bestp >= lmax) break;
        if (lane == 0) {
          int p = PS - 2;                        // insert, drop old max
          while (p >= 0 && list[goff + p] > bestp) {
            list[goff + p + 1] = list[goff + p];
            --p;
          }
          list[goff + p + 1] = bestp;
          const int nloc = (int)(bestp & 0xFFFFFFFFull) - pbase;
          skey[g * CHUNK + nloc] = 0xFFFFFFFFu;  // exclude this point
        }
        __builtin_amdgcn_wave_barrier();
      }
    }
    __syncthreads();
  }

  // ---- emit p = nn - c, ascending by (distance, index) ----
  for (int t = tid; t < CT * PS; t += 256) {
    const int g = t >> 6, j = t & 63;
    const unsigned long long pk = list[g * PS + j];
    const int idx = (int)(unsigned)(pk & 0xFFFFFFFFull);
    const int gc  = cbase + g;
    const float* cc = c_in + ((size_t)b * KC + gc) * 3;
    const float* pp = xb + (size_t)idx * 3;
    float* oo = p_out + (((size_t)b * KC + gc) * PS + j) * 3;
    oo[0] = pp[0] - cc[0];
    oo[1] = pp[1] - cc[1];
    oo[2] = pp[2] - cc[2];
  }
}

extern "C" void kernel_launch(void* const* d_in, const int* in_sizes, int n_in,
                              void* d_out, int out_size, void* d_ws, size_t ws_size,
                              hipStream_t stream) {
  (void)in_sizes; (void)n_in; (void)d_ws; (void)ws_size; (void)out_size;
  const float* x = (const float*)d_in[0];
  float* out   = (float*)d_out;
  float* p_out = out;                                   // [32][256][64][3]
  float* c_out = out + (size_t)BATCH * KC * PS * 3;     // [32][256][3]

  fps_kernel<<<BATCH, 1024, 0, stream>>>(x, c_out);
  knn_kernel<<<BATCH * (KC / CT), 256, 0, stream>>>(x, c_out, p_out);
}